// DirectionMinGlobalNodeLoss_67362267070813
// MI455X (gfx1250) — compile-verified
//
#include <hip/hip_runtime.h>

typedef __attribute__((ext_vector_type(2))) float v2f;
typedef __attribute__((ext_vector_type(8))) float v8f;

namespace {
constexpr int   kB       = 64;      // graphs
constexpr int   kAtoms   = 16384;   // atoms per graph
constexpr int   kG       = 8;       // global nodes per graph
constexpr int   kBPG     = 8;       // blocks per graph
constexpr int   kThreads = 256;     // 8 waves (wave32)
constexpr int   kWaves   = kThreads / 32;
constexpr float kInvEps  = 1e8f;    // 1 / EPS, EPS = 1e-8
}

__global__ void zero_ws_kernel(float* ws, int n) {
  int i = blockIdx.x * blockDim.x + threadIdx.x;
  if (i < n) ws[i] = 0.0f;
}

// Per-block partial column sums of cos(tdir_i, gpos_g - atom_i).
// grid = kB * kBPG blocks, 256 threads. ws[b*kG + g] accumulates sum over atoms.
__global__ __launch_bounds__(kThreads)
void cos_partial_kernel(const float* __restrict__ pos,
                        const float* __restrict__ gpos,
                        const float* __restrict__ tdir,
                        float* __restrict__ ws)
{
  const int b    = blockIdx.x / kBPG;
  const int blk  = blockIdx.x % kBPG;
  const int lane = threadIdx.x & 31;
  const int wave = threadIdx.x >> 5;
  const int n    = lane & 15;         // column index in 16x16 D tile
  const bool hi  = lane >= 16;

  __shared__ float colsum[kG];
  if (threadIdx.x < kG) colsum[threadIdx.x] = 0.0f;
  __syncthreads();

  // B-matrix (4x16 f32): VGPR0/1 lanes 0-15 hold rows K=0,1; lanes 16-31 rows K=2,3.
  // Columns 0..7 = the 8 global nodes of this graph; 8..15 = zero (unused).
  v2f bmat; bmat.x = 0.0f; bmat.y = 0.0f;
  float c2 = 0.0f;                     // |gpos_n|^2 for this lane's column
  if (n < kG) {
    const float* gp = gpos + (size_t)(b * kG + n) * 3;
    float gx = gp[0], gy = gp[1], gz = gp[2];
    c2 = gx * gx + gy * gy + gz * gz;
    bmat.x = hi ? gz : gx;
    bmat.y = hi ? 0.0f : gy;
  }

  float acc[8];
  #pragma unroll
  for (int r = 0; r < 8; ++r) acc[r] = 0.0f;

  const int atomsPerBlock = kAtoms / kBPG;              // 2048
  const int iters = atomsPerBlock / (kWaves * 16);      // 16
  const int blockBase = b * kAtoms + blk * atomsPerBlock;

  for (int it = 0; it < iters; ++it) {
    const int i0 = blockBase + it * (kWaves * 16) + wave * 16;
    const int ai = i0 + n;            // atom handled by this lane's A-matrix slot
    v2f aT, aP;                       // A-matrices: tdir and atom positions
    float sv = 0.0f, t2v = 0.0f, a2v = 0.0f;  // per-atom scalars (lanes 0-15)
    if (!hi) {
      float tx = tdir[ai * 3 + 0], ty = tdir[ai * 3 + 1], tz = tdir[ai * 3 + 2];
      float px = pos [ai * 3 + 0], py = pos [ai * 3 + 1], pz = pos [ai * 3 + 2];
      aT.x = tx; aT.y = ty;
      aP.x = px; aP.y = py;
      sv  = tx * px + ty * py + tz * pz;   // tdir . atom
      t2v = tx * tx + ty * ty + tz * tz;   // |tdir|^2
      a2v = px * px + py * py + pz * pz;   // |atom|^2
    } else {
      aT.x = tdir[ai * 3 + 2]; aT.y = 0.0f;  // K=2 (z), K=3 zero pad
      aP.x = pos [ai * 3 + 2]; aP.y = 0.0f;
    }

    v8f zero = {};
    // TG = tdir(16x4) x gpos(4x16); AG = atom(16x4) x gpos(4x16)
    v8f TG = __builtin_amdgcn_wmma_f32_16x16x4_f32(
        false, aT, false, bmat, (short)0, zero, false, false);
    v8f AG = __builtin_amdgcn_wmma_f32_16x16x4_f32(
        false, aP, false, bmat, (short)0, zero, false, false);

    #pragma unroll
    for (int r = 0; r < 8; ++r) {
      const int m = r + (hi ? 8 : 0);        // D-tile row (atom index within 16)
      float sM  = __shfl(sv,  m, 32);
      float t2M = __shfl(t2v, m, 32);
      float a2M = __shfl(a2v, m, 32);
      float num = TG[r] - sM;                            // tdir.(gpos - atom)
      float p2  = fmaxf(c2 - 2.0f * AG[r] + a2M, 0.0f);  // |gpos - atom|^2
      // cos = num / max(sqrt(t2*p2), eps) == num * min(rsqrt(t2*p2), 1/eps)
      float rin = __builtin_amdgcn_rsqf(t2M * p2);       // v_rsq_f32 (TRANS)
      acc[r] += num * fminf(rin, kInvEps);
    }
  }

  float sum8 = 0.0f;
  #pragma unroll
  for (int r = 0; r < 8; ++r) sum8 += acc[r];

  // Column n partials live in lanes n and n+16 of every wave; only n<8 valid.
  if (n < kG) atomicAdd(&colsum[n], sum8);   // ds_add_f32
  __syncthreads();
  if (threadIdx.x < kG)
    atomicAdd(&ws[b * kG + threadIdx.x], colsum[threadIdx.x]);
}

// One block of kB threads: per-graph min/argmin over g, then mean over graphs.
__global__ __launch_bounds__(kB)
void finalize_kernel(const float* __restrict__ ws, float* __restrict__ out)
{
  __shared__ float red[kB];
  const int b = threadIdx.x;
  float minv = 3.402823e38f;
  int mini = 0;
  #pragma unroll
  for (int g = 0; g < kG; ++g) {
    float loss = 1.0f - ws[b * kG + g] * (1.0f / (float)kAtoms);
    if (loss < minv) { minv = loss; mini = g; }   // first-min, matches argmin
  }
  out[1 + b] = (float)mini;
  red[b] = minv;
  __syncthreads();
  for (int s = kB / 2; s > 0; s >>= 1) {
    if (b < s) red[b] += red[b + s];
    __syncthreads();
  }
  if (b == 0) out[0] = red[0] * (1.0f / (float)kB);
}

extern "C" void kernel_launch(void* const* d_in, const int* in_sizes, int n_in,
                              void* d_out, int out_size, void* d_ws, size_t ws_size,
                              hipStream_t stream) {
  (void)in_sizes; (void)n_in; (void)out_size; (void)ws_size;
  const float* pos  = (const float*)d_in[0];  // atom_positions [B*N,3]
  const float* gpos = (const float*)d_in[1];  // pred_pos_global_node [B*G,3]
  const float* tdir = (const float*)d_in[2];  // true_direction_vectors [B*N,3]
  float* ws  = (float*)d_ws;
  float* out = (float*)d_out;

  const int wsN = kB * kG;
  zero_ws_kernel<<<(wsN + 255) / 256, 256, 0, stream>>>(ws, wsN);
  cos_partial_kernel<<<kB * kBPG, kThreads, 0, stream>>>(pos, gpos, tdir, ws);
  finalize_kernel<<<1, kB, 0, stream>>>(ws, out);
}